// MultiTaskLoss_65910568124972
// MI455X (gfx1250) — compile-verified
//
#include <hip/hip_runtime.h>
#include <hip/hip_bf16.h>

#define NB   8
#define NA   200000
#define NM   64
#define BLK  256
#define NBLK ((NA + BLK - 1) / BLK)   // 782 blocks per image

typedef __attribute__((ext_vector_type(2))) float v2f;
typedef __attribute__((ext_vector_type(8))) float v8f;

struct SelStats {               // 24 bytes, all zero-initialized
    unsigned pos_cnt;
    unsigned neg_cnt;
    unsigned eff;
    unsigned k_rem;
    unsigned prefix;
    unsigned pad;
};

// ---------------------------------------------------------------- utilities
__device__ __forceinline__ unsigned float_sortkey(float f) {
    unsigned bits = __float_as_uint(f);
    return (bits & 0x80000000u) ? ~bits : (bits | 0x80000000u);
}
__device__ __forceinline__ float sortkey_float(unsigned key) {
    unsigned bits = (key & 0x80000000u) ? (key & 0x7FFFFFFFu) : ~key;
    return __uint_as_float(bits);
}

// Exact f32 wave32 reduction via V_WMMA_F32_16X16X4_F32.
// A(16x4): lane m vgpr{0,1} = A[m][{0,1}], lane m+16 vgpr{0,1} = A[m][{2,3}].
// With a.x = lane partial, a.y = 0 and B = ones: D[m][n] = s_m + s_{m+16}.
// Each lane sums its 8 D vgprs -> lanes 0-15 hold sum(m=0..7), lanes 16-31
// hold sum(m=8..15); one LDS exchange completes the total.
__device__ __forceinline__ float wave_sum_wmma(float v, float* xch) {
    v2f a;  a.x = v;    a.y = 0.0f;
    v2f bm; bm.x = 1.0f; bm.y = 1.0f;
    v8f c = {};
    v8f d = __builtin_amdgcn_wmma_f32_16x16x4_f32(
        false, a, false, bm, (short)0, c, false, false);
    float s = d[0] + d[1] + d[2] + d[3] + d[4] + d[5] + d[6] + d[7];
    if (threadIdx.x == 0)  xch[0] = s;
    if (threadIdx.x == 16) xch[1] = s;
    __syncthreads();
    float r = xch[0] + xch[1];
    __syncthreads();
    return r;
}

// ---------------------------------------------------------------- kernels
__global__ void zero_u32(unsigned* __restrict__ p, int n) {
    int i = blockIdx.x * blockDim.x + threadIdx.x;
    if (i < n) p[i] = 0u;
}

// Pass 1: per-anchor IoU max/argmax over 64 boxes (boxes staged in LDS),
// pos/neg flags, pos cls sum, smooth-L1 box sum, sortable neg key.
__global__ __launch_bounds__(BLK) void pass1(
    const float* __restrict__ cls,   // [B][A][2]
    const float* __restrict__ reg,   // [B][A][4]
    const float* __restrict__ anc,   // [1][A][4]
    const float* __restrict__ ann,   // [B][M][4]
    unsigned*    __restrict__ keys,  // [B][A]
    SelStats*    __restrict__ stats,
    float*       __restrict__ pos_part,  // [B][NBLK]
    float*       __restrict__ box_part)  // [B][NBLK]
{
    __shared__ float4 sbox[NM];
    __shared__ float  sarea[NM];
    __shared__ float  red[BLK];
    __shared__ int    ired[BLK];

    const int b = blockIdx.y;
    const int a = blockIdx.x * BLK + threadIdx.x;

    if (threadIdx.x < NM * 4)   // BLK == 256 == NM*4
        ((float*)sbox)[threadIdx.x] = ann[(size_t)b * NM * 4 + threadIdx.x];
    __syncthreads();
    if (threadIdx.x < NM) {
        float4 bb = sbox[threadIdx.x];
        sarea[threadIdx.x] = (bb.z - bb.x) * (bb.w - bb.y);
    }
    __syncthreads();

    float posv = 0.0f, boxv = 0.0f;
    int   isPos = 0, isNeg = 0;

    if (a < NA) {
        float4 ab = *(const float4*)(anc + (size_t)a * 4);
        float aw = ab.z - ab.x, ah = ab.w - ab.y;
        float area_a = aw * ah;

        float best = -__builtin_inff();
        int   argj = 0;
#pragma unroll 4
        for (int j = 0; j < NM; ++j) {
            float4 bb = sbox[j];
            float iw = fmaxf(fminf(ab.z, bb.z) - fmaxf(ab.x, bb.x), 0.0f);
            float ih = fmaxf(fminf(ab.w, bb.w) - fmaxf(ab.y, bb.y), 0.0f);
            float inter = iw * ih;
            float ua = fmaxf(area_a + sarea[j] - inter, 1e-8f);
            float iou = inter / ua;
            float v = (bb.x > 0.0f) ? iou : -1.0f;   // validity mask
            if (v > best) { best = v; argj = j; }    // first-max like argmax
        }
        isPos = (best >= 0.5f);
        isNeg = (best < 0.3f);

        float2 c = *(const float2*)(cls + ((size_t)b * NA + a) * 2);
        if (isPos) {
            posv = -c.x;
            float4 g = sbox[argj];
            float gw = g.z - g.x, gh = g.w - g.y;
            float gcx = g.x + 0.5f * gw, gcy = g.y + 0.5f * gh;
            float acx = ab.x + 0.5f * aw, acy = ab.y + 0.5f * ah;
            float dx = ((gcx - acx) / (aw + 1e-14f)) / 0.1f;
            float dy = ((gcy - acy) / (ah + 1e-14f)) / 0.1f;
            float dw = logf(gw / aw) / 0.2f;
            float dh = logf(gh / ah) / 0.2f;
            float4 r = *(const float4*)(reg + ((size_t)b * NA + a) * 4);
            float t0 = fabsf(dx - r.x), t1 = fabsf(dy - r.y);
            float t2 = fabsf(dw - r.z), t3 = fabsf(dh - r.w);
            boxv  = (t0 < 1.0f) ? 0.5f * t0 * t0 : t0 - 0.5f;
            boxv += (t1 < 1.0f) ? 0.5f * t1 * t1 : t1 - 0.5f;
            boxv += (t2 < 1.0f) ? 0.5f * t2 * t2 : t2 - 0.5f;
            boxv += (t3 < 1.0f) ? 0.5f * t3 * t3 : t3 - 0.5f;
        }
        keys[(size_t)b * NA + a] = isNeg ? float_sortkey(-c.y) : 0u;
    }

    // deterministic block reductions
    red[threadIdx.x] = posv; __syncthreads();
    for (int s = BLK / 2; s > 0; s >>= 1) {
        if (threadIdx.x < s) red[threadIdx.x] += red[threadIdx.x + s];
        __syncthreads();
    }
    if (threadIdx.x == 0) pos_part[b * NBLK + blockIdx.x] = red[0];
    __syncthreads();

    red[threadIdx.x] = boxv; __syncthreads();
    for (int s = BLK / 2; s > 0; s >>= 1) {
        if (threadIdx.x < s) red[threadIdx.x] += red[threadIdx.x + s];
        __syncthreads();
    }
    if (threadIdx.x == 0) box_part[b * NBLK + blockIdx.x] = red[0];
    __syncthreads();

    ired[threadIdx.x] = isPos; __syncthreads();
    for (int s = BLK / 2; s > 0; s >>= 1) {
        if (threadIdx.x < s) ired[threadIdx.x] += ired[threadIdx.x + s];
        __syncthreads();
    }
    if (threadIdx.x == 0 && ired[0] > 0)
        atomicAdd(&stats[b].pos_cnt, (unsigned)ired[0]);
    __syncthreads();

    ired[threadIdx.x] = isNeg; __syncthreads();
    for (int s = BLK / 2; s > 0; s >>= 1) {
        if (threadIdx.x < s) ired[threadIdx.x] += ired[threadIdx.x + s];
        __syncthreads();
    }
    if (threadIdx.x == 0 && ired[0] > 0)
        atomicAdd(&stats[b].neg_cnt, (unsigned)ired[0]);
}

// Radix-select histogram pass over neg keys matching the current prefix.
__global__ __launch_bounds__(BLK) void hist_pass(
    const unsigned* __restrict__ keys, const SelStats* __restrict__ stats,
    unsigned* __restrict__ hist, unsigned prefix_mask, int shift, int nbins)
{
    __shared__ unsigned h[2048];
    const int b = blockIdx.y;
    for (int i = threadIdx.x; i < nbins; i += BLK) h[i] = 0u;
    __syncthreads();

    const unsigned prefix = stats[b].prefix;
    const int a = blockIdx.x * BLK + threadIdx.x;
    if (a < NA) {
        unsigned key = keys[(size_t)b * NA + a];
        if (key != 0u && (key & prefix_mask) == prefix)
            atomicAdd(&h[(key >> shift) & (unsigned)(nbins - 1)], 1u);
    }
    __syncthreads();
    for (int i = threadIdx.x; i < nbins; i += BLK) {
        unsigned v = h[i];
        if (v) atomicAdd(&hist[b * 2048 + i], v);
    }
}

// Scan bins high->low to locate the bin containing the k-th largest key;
// refine prefix/k_rem. Level 0 computes eff = min(3*num_pos, num_neg).
__global__ __launch_bounds__(BLK) void scan_pass(
    unsigned* __restrict__ hist, SelStats* __restrict__ stats,
    int shift, int nbins, int level)
{
    __shared__ unsigned h[2048];
    const int b = blockIdx.x;
    for (int i = threadIdx.x; i < nbins; i += BLK) {
        h[i] = hist[b * 2048 + i];
        hist[b * 2048 + i] = 0u;   // reset for next level
    }
    __syncthreads();

    if (threadIdx.x == 0) {
        SelStats* st = &stats[b];
        unsigned k;
        if (level == 0) {
            unsigned np = st->pos_cnt, nn = st->neg_cnt;
            unsigned eff = (3u * np < nn) ? 3u * np : nn;
            st->eff = eff;
            if (eff == 0u) { st->prefix = 0xFFFFFFFFu; st->k_rem = 0u; return; }
            k = eff;
        } else {
            k = st->k_rem;
            if (k == 0u) return;
        }
        unsigned cum = 0u;
        int bin = 0;
        for (int i = nbins - 1; i >= 0; --i) {
            unsigned c = h[i];
            if (cum + c >= k) { bin = i; break; }
            cum += c;
        }
        st->k_rem   = k - cum;
        st->prefix |= ((unsigned)bin) << shift;
    }
}

// Sum of all neg values strictly greater than the selected threshold key.
__global__ __launch_bounds__(BLK) void sum_gt_pass(
    const unsigned* __restrict__ keys, const SelStats* __restrict__ stats,
    float* __restrict__ gt_part)
{
    __shared__ float red[BLK];
    const int b = blockIdx.y;
    const unsigned t = stats[b].prefix;
    const int a = blockIdx.x * BLK + threadIdx.x;
    float v = 0.0f;
    if (a < NA) {
        unsigned key = keys[(size_t)b * NA + a];
        if (key != 0u && key > t) v = sortkey_float(key);
    }
    red[threadIdx.x] = v; __syncthreads();
    for (int s = BLK / 2; s > 0; s >>= 1) {
        if (threadIdx.x < s) red[threadIdx.x] += red[threadIdx.x + s];
        __syncthreads();
    }
    if (threadIdx.x == 0) gt_part[b * NBLK + blockIdx.x] = red[0];
}

// Final fixed-order reduction (WMMA-based wave32 sum) + loss assembly.
__global__ __launch_bounds__(32) void finalize(
    const float* __restrict__ pos_part, const float* __restrict__ box_part,
    const float* __restrict__ gt_part,  const SelStats* __restrict__ stats,
    float* __restrict__ out)
{
    __shared__ float xch[2];
    const int b = blockIdx.x;
    const int lane = threadIdx.x;

    float ps = 0.0f, bs = 0.0f, gs = 0.0f;
    for (int i = lane; i < NBLK; i += 32) {
        ps += pos_part[b * NBLK + i];
        bs += box_part[b * NBLK + i];
        gs += gt_part[b * NBLK + i];
    }
    float pst = wave_sum_wmma(ps, xch);
    float bst = wave_sum_wmma(bs, xch);
    float gst = wave_sum_wmma(gs, xch);

    if (lane == 0) {
        SelStats st = stats[b];
        float tval = (st.eff > 0u) ? sortkey_float(st.prefix) : 0.0f;
        float neg_sum = gst + (float)st.k_rem * tval;
        float efff  = fmaxf((float)st.eff, 1.0f);
        float nposf = fmaxf((float)st.pos_cnt, 1.0f);
        float cls_loss = (st.pos_cnt > 0u) ? (pst / nposf + neg_sum / efff) : 0.0f;
        float denom = fmaxf((float)(st.pos_cnt * 4u), 1.0f);
        float box_loss = (st.pos_cnt > 0u) ? (bst / denom) : 0.0f;
        out[b]     = cls_loss;
        out[8 + b] = box_loss;
    }
}

// ---------------------------------------------------------------- launcher
extern "C" void kernel_launch(void* const* d_in, const int* in_sizes, int n_in,
                              void* d_out, int out_size, void* d_ws, size_t ws_size,
                              hipStream_t stream) {
    const float* cls = (const float*)d_in[0];   // [8][200000][2]
    const float* reg = (const float*)d_in[1];   // [8][200000][4]
    const float* anc = (const float*)d_in[2];   // [1][200000][4]
    const float* ann = (const float*)d_in[3];   // [8][64][4]
    float* out = (float*)d_out;                 // 8 cls losses then 8 box losses

    // workspace layout (~6.6 MB)
    char* p = (char*)d_ws;
    unsigned* keys = (unsigned*)p;  p += (size_t)NB * NA * sizeof(unsigned);
    unsigned* hist = (unsigned*)p;  p += (size_t)NB * 2048 * sizeof(unsigned);
    SelStats* st   = (SelStats*)p;  p += (size_t)NB * sizeof(SelStats);
    float* pos_part = (float*)p;    p += (size_t)NB * NBLK * sizeof(float);
    float* box_part = (float*)p;    p += (size_t)NB * NBLK * sizeof(float);
    float* gt_part  = (float*)p;

    // zero hist + stats (contiguous)
    int zcnt = NB * 2048 + NB * (int)(sizeof(SelStats) / 4);
    zero_u32<<<(zcnt + 255) / 256, 256, 0, stream>>>(hist, zcnt);

    dim3 g(NBLK, NB);
    pass1<<<g, BLK, 0, stream>>>(cls, reg, anc, ann, keys, st, pos_part, box_part);

    // 3-level radix select: bits [31:21], [20:10], [9:0]
    hist_pass<<<g, BLK, 0, stream>>>(keys, st, hist, 0x00000000u, 21, 2048);
    scan_pass<<<NB, BLK, 0, stream>>>(hist, st, 21, 2048, 0);
    hist_pass<<<g, BLK, 0, stream>>>(keys, st, hist, 0xFFE00000u, 10, 2048);
    scan_pass<<<NB, BLK, 0, stream>>>(hist, st, 10, 2048, 1);
    hist_pass<<<g, BLK, 0, stream>>>(keys, st, hist, 0xFFFFFC00u, 0, 1024);
    scan_pass<<<NB, BLK, 0, stream>>>(hist, st, 0, 1024, 2);

    sum_gt_pass<<<g, BLK, 0, stream>>>(keys, st, gt_part);
    finalize<<<NB, 32, 0, stream>>>(pos_part, box_part, gt_part, st, out);
}